// ConvTransform_88055419503065
// MI455X (gfx1250) — compile-verified
//
#include <hip/hip_runtime.h>
#include <stdint.h>

typedef float    v2f  __attribute__((ext_vector_type(2)));
typedef float    v8f  __attribute__((ext_vector_type(8)));
typedef uint32_t u32x4 __attribute__((ext_vector_type(4)));
typedef int      i32x4 __attribute__((ext_vector_type(4)));
typedef int      i32x8 __attribute__((ext_vector_type(8)));

#define LDS_STRIDE 72   // 64 data cols + 8 zero-gap cols (TDM pad), 288 B rows
#define LDS_ROWS   66   // row 0 and row 65 are zero halo rows

// Depthwise 3x3 conv over 8192 independent 64x64 fp32 planes.
// One workgroup (8 waves) per plane. TDM loads the plane into LDS with
// padded row stride; compute is 12x v_wmma_f32_16x16x4_f32 per 16x16 tile
// (tridiagonal band-matrix formulation of the horizontal conv, accumulated
// over the 3 vertical taps) + masked VALU edge correction.
__global__ __launch_bounds__(256)
void dwconv3x3_wmma_kernel(const float* __restrict__ x,
                           const float* __restrict__ Wt,
                           float* __restrict__ out)
{
    __shared__ float smem[LDS_ROWS * LDS_STRIDE];

    const int plane = blockIdx.x;      // (n*C + c), 8192 total
    const int c     = plane & 255;     // C = 256
    const int tid   = threadIdx.x;
    const int lane  = tid & 31;
    const int wave  = tid >> 5;

    // ---- zero halo cells: rows 0 and 65 fully, gap cols 64..71 of rows 1..64
    for (int i = tid; i < 2 * LDS_STRIDE; i += 256) {
        int r   = (i < LDS_STRIDE) ? 0 : (LDS_ROWS - 1);
        int col = (i < LDS_STRIDE) ? i : (i - LDS_STRIDE);
        smem[r * LDS_STRIDE + col] = 0.0f;
    }
    for (int i = tid; i < 64 * 8; i += 256) {
        int r   = 1 + (i >> 3);
        int col = 64 + (i & 7);
        smem[r * LDS_STRIDE + col] = 0.0f;
    }

    // ---- TDM: DMA the 64x64 fp32 plane into LDS rows 1..64, cols 0..63,
    //      with 8-dword padding after every 64 dwords -> row stride 72 floats.
    const uint64_t gaddr = (uint64_t)(uintptr_t)(x + (size_t)plane * 4096);
    if (wave == 0) {
        uint32_t lds0 = (uint32_t)(uintptr_t)(&smem[0]) + LDS_STRIDE * 4; // row 1
        u32x4 g0;
        g0[0] = 1u;                                   // count=1, user descriptor
        g0[1] = lds0;                                 // lds_addr (bytes)
        g0[2] = (uint32_t)(gaddr & 0xFFFFFFFFu);      // global_addr[31:0]
        g0[3] = (uint32_t)((gaddr >> 32) & 0x01FFFFFFu) | (2u << 30); // addr hi | type=2
        i32x8 g1;
        g1[0] = (2 << 16)                             // data_size = 4 B
              | (1 << 20)                             // pad_enable
              | (5 << 22)                             // pad_interval = 64 dwords
              | (7 << 25);                            // pad_amount  = 8 dwords
        g1[1] = (64 << 16);                           // tensor_dim0[15:0]
        g1[2] = (64 << 16);                           // tensor_dim0 hi=0 | tensor_dim1 lo=64
        g1[3] = (64 << 16);                           // tensor_dim1 hi=0 | tile_dim0=64
        g1[4] = 64;                                   // tile_dim1=64 | tile_dim2=0
        g1[5] = 64;                                   // tensor_dim0_stride lo = 64
        g1[6] = (4096 << 16);                         // stride0 hi=0 | tensor_dim1_stride lo
        g1[7] = 0;                                    // tensor_dim1_stride hi
        i32x4 gz4 = {0, 0, 0, 0};
        i32x8 gz8 = {0, 0, 0, 0, 0, 0, 0, 0};
        __builtin_amdgcn_tensor_load_to_lds(g0, g1, gz4, gz4, gz8, 0);
        __builtin_amdgcn_s_wait_tensorcnt(0);
    }
    __syncthreads();

    // ---- per-channel 3x3 taps (uniform within workgroup -> scalar loads)
    const float* wp = Wt + ((size_t)c * 256 + c) * 9;
    const float taps[3][3] = {{wp[0], wp[1], wp[2]},
                              {wp[3], wp[4], wp[5]},
                              {wp[6], wp[7], wp[8]}};

    const int n    = lane & 15;   // output column within tile / A-matrix M row
    const int half = lane >> 4;

    // ---- build tridiagonal band matrices B_kh (4x16 sub-blocks, K-chained x4)
    // B[k][n] = taps[kh][k-n+1] for |k-n|<=1 else 0.
    // VGPR pair layout: .x -> klocal=2*half, .y -> klocal=2*half+1.
    v2f Bm[3][4];
    #pragma unroll
    for (int kh = 0; kh < 3; ++kh) {
        #pragma unroll
        for (int kk = 0; kk < 4; ++kk) {
            int k0 = 4 * kk + 2 * half;
            int k1 = k0 + 1;
            float b0 = (k0 == n - 1) ? taps[kh][0]
                     : (k0 == n    ) ? taps[kh][1]
                     : (k0 == n + 1) ? taps[kh][2] : 0.0f;
            float b1 = (k1 == n - 1) ? taps[kh][0]
                     : (k1 == n    ) ? taps[kh][1]
                     : (k1 == n + 1) ? taps[kh][2] : 0.0f;
            v2f b; b.x = b0; b.y = b1;
            Bm[kh][kk] = b;
        }
    }

    // ---- each wave: 16-row strip, two 16-col tiles
    const int h0 = (wave >> 1) * 16;
    const int ctbase = (wave & 1) * 2;
    float* outp = out + (size_t)plane * 4096;

    #pragma unroll
    for (int t = 0; t < 2; ++t) {
        const int c0 = (ctbase + t) * 16;
        v8f d = {0.0f, 0.0f, 0.0f, 0.0f, 0.0f, 0.0f, 0.0f, 0.0f};

        #pragma unroll
        for (int kh = 0; kh < 3; ++kh) {
            // A[m][k] = xpad[h0+m+kh-1][c0+k]; lane m = n; lds row = xpad row + 1
            const int rowbase = (h0 + n + kh) * LDS_STRIDE;
            #pragma unroll
            for (int kk = 0; kk < 4; ++kk) {
                const int col = c0 + 4 * kk + 2 * half;
                v2f a;
                a.x = smem[rowbase + col];
                a.y = smem[rowbase + col + 1];
                d = __builtin_amdgcn_wmma_f32_16x16x4_f32(
                        false, a, false, Bm[kh][kk], (short)0, d, false, false);
            }
        }

        // ---- edge corrections: out col 0 needs x[.][c0-1], col 15 needs x[.][c0+16].
        // c0==0 -> col 71 (zero gap), c0==48 -> col 64 (zero gap): halo is free.
        const int colL  = (c0 == 0) ? (LDS_STRIDE - 1) : (c0 - 1);
        const int colR  = c0 + 16;
        const int ecol  = (n == 15) ? colR : colL;
        const float emask = (n == 0 || n == 15) ? 1.0f : 0.0f;

        #pragma unroll
        for (int v = 0; v < 8; ++v) {
            const int m = v + 8 * half;     // D layout: VGPR v holds M=v / M=v+8
            float corr = 0.0f;
            #pragma unroll
            for (int kh = 0; kh < 3; ++kh) {
                const float tap = (n == 15) ? taps[kh][2] : taps[kh][0];
                corr += tap * smem[(h0 + m + kh) * LDS_STRIDE + ecol];
            }
            float val = d[v] + emask * corr;
            outp[(h0 + m) * 64 + c0 + n] = val;
        }
    }
}

extern "C" void kernel_launch(void* const* d_in, const int* in_sizes, int n_in,
                              void* d_out, int out_size, void* d_ws, size_t ws_size,
                              hipStream_t stream) {
    const float* x  = (const float*)d_in[0];   // (2,16,256,64,64) fp32
    const float* Wm = (const float*)d_in[1];   // (256,256,3,3)    fp32
    float* out = (float*)d_out;                // same shape as x

    dim3 grid(8192);   // one workgroup per (n, c) plane
    dim3 block(256);   // 8 waves (wave32)
    hipLaunchKernelGGL(dwconv3x3_wmma_kernel, grid, block, 0, stream, x, Wm, out);
}